// MultiPINNNetwork_87162066305743
// MI455X (gfx1250) — compile-verified
//
#include <hip/hip_runtime.h>
#include <math.h>

typedef __attribute__((ext_vector_type(2))) float v2f;
typedef __attribute__((ext_vector_type(8))) float v8f;

#define TT 4096
#define BB 256

__device__ __forceinline__ float fast_tanh(float x) {
    // tanh(x) = 1 - 2/(exp(2x)+1); exp->v_exp_f32, rcp->v_rcp_f32
    float e = __expf(2.0f * x);
    return 1.0f - 2.0f * __builtin_amdgcn_rcpf(e + 1.0f);
}

__device__ __forceinline__ v8f wmma4(v2f a, v2f b, v8f c) {
    // V_WMMA_F32_16X16X4_F32 : D = A(16x4,f32) * B(4x16,f32) + C(16x16,f32)
    return __builtin_amdgcn_wmma_f32_16x16x4_f32(
        false, a, false, b, (short)0, c, false, false);
}

__global__ __launch_bounds__(512)
void pinn_kernel(const float* __restrict__ x,
                 const float* __restrict__ W1,  const float* __restrict__ b1,
                 const float* __restrict__ W2,  const float* __restrict__ b2,
                 const float* __restrict__ W3,  const float* __restrict__ b3,
                 const float* __restrict__ logE,const float* __restrict__ logeta,
                 const float* __restrict__ imp, const float* __restrict__ lw,
                 const float* __restrict__ Wf1, const float* __restrict__ bf1,
                 const float* __restrict__ Wf2, const float* __restrict__ bf2,
                 float* __restrict__ out)
{
    // per-wave private scratch (padded strides: conflict-free banks)
    __shared__ __align__(16) float h1s[16][16 * 34];   // 34.8 KB
    __shared__ __align__(16) float h2s[16][16 * 20];   // 20.5 KB
    __shared__ __align__(16) float sbuf[2][16][16];    // [buf][neuron][m]

    const int tid  = threadIdx.x;
    const int n    = tid >> 5;          // wave id == neuron id == head batch-in-tile
    const int lane = tid & 31;
    const int hi   = lane >> 4;         // half of wave
    const int lm   = lane & 15;
    const int b0   = blockIdx.x * 16;
    const int kb   = hi ? 2 : 0;

    // ---- loop-invariant per-neuron state -------------------------------
    const float E   = __expf(logE[n]);
    const float eta = __expf(logeta[n]);
    const float iw  = 1.0f / (1.0f + __expf(-imp[n]));
    const float b3v = b3[n];

    // layer1 weights (B-layout, 2 column halves), bias in C-layout
    v2f B1[2]; v8f C1[2];
#pragma unroll
    for (int h = 0; h < 2; ++h) {
        int j = 16 * h + lm;
        B1[h].x = W1[(n * 32 + j) * 3 + kb];                 // k = 0 or 2
        B1[h].y = hi ? 0.0f : W1[(n * 32 + j) * 3 + 1];      // k = 1, pad k=3
        float bb = b1[n * 32 + j];
#pragma unroll
        for (int i = 0; i < 8; ++i) C1[h][i] = bb;
    }
    // layer2 weights: 8 K-chunks of 4 in B-layout
    v2f B2[8];
#pragma unroll
    for (int kc = 0; kc < 8; ++kc) {
        B2[kc].x = W2[(n * 16 + lm) * 32 + 4 * kc + kb];
        B2[kc].y = W2[(n * 16 + lm) * 32 + 4 * kc + kb + 1];
    }
    const float c2s = b2[n * 16 + lm];

    // layer3 weights (wave-uniform -> scalar regs)
    float w3v[16];
#pragma unroll
    for (int k = 0; k < 16; ++k) w3v[k] = W3[n * 16 + k];

    // head constants
    float cl[4];
    {
        float e0 = __expf(lw[0]), e1 = __expf(lw[1]);
        float e2 = __expf(lw[2]), e3 = __expf(lw[3]);
        float es = e0 + e1 + e2 + e3;
        float ws[4] = { e0 / es, e1 / es, e2 / es, e3 / es };
#pragma unroll
        for (int l = 0; l < 4; ++l) {
            float s = 0.0f;
#pragma unroll
            for (int p = 0; p < 4; ++p)
                s += 1.0f / (1.0f + __expf(-imp[l * 4 + p]));
            cl[l] = (s / (s + 1e-8f)) * ws[l];
        }
    }
    float wf1v[4];
#pragma unroll
    for (int l = 0; l < 4; ++l) wf1v[l] = Wf1[lm * 4 + l];
    const float bf1v = bf1[lm];
    const float wf2v = Wf2[lm];
    const float bf2v = bf2[0];

    // ---- recurrence ----------------------------------------------------
    float prev = 0.0f;
    const float2* xp = (const float2*)x;         // x[b][t] as float2
    float2 xc = make_float2(0.0f, 0.0f);
    if (!hi) xc = xp[(size_t)(b0 + lm) * TT];

    float* h1p = h1s[n];
    float* h2p = h2s[n];

    for (int t = 0; t < TT; ++t) {
        // prefetch next x
        float2 xn = make_float2(0.0f, 0.0f);
        if (!hi && (t + 1) < TT) xn = xp[(size_t)(b0 + lm) * TT + (t + 1)];

        // A1: lanes0-15 hold (strain, rate) [K=0,1]; lanes16-31 (prev, 0) [K=2,3]
        v2f a1;
        a1.x = hi ? prev : xc.x;
        a1.y = hi ? 0.0f : xc.y;

        // ---- layer 1: 16x32 = A(16x4) x W1^T(4x32) ----
        v8f d0 = wmma4(a1, B1[0], C1[0]);
        v8f d1 = wmma4(a1, B1[1], C1[1]);
#pragma unroll
        for (int i = 0; i < 8; ++i) { d0[i] = fast_tanh(d0[i]); d1[i] = fast_tanh(d1[i]); }

        // scatter h1 (D-layout -> row-major LDS, stride 34)
#pragma unroll
        for (int v = 0; v < 8; ++v) {
            h1p[(v + 8 * hi) * 34 + lm]      = d0[v];
            h1p[(v + 8 * hi) * 34 + 16 + lm] = d1[v];
        }

        // ---- layer 2: 8 chained K=4 WMMAs, bias preloaded in C ----
        v8f acc;
#pragma unroll
        for (int i = 0; i < 8; ++i) acc[i] = c2s;
#pragma unroll
        for (int kc = 0; kc < 8; ++kc) {
            v2f a2 = *(const v2f*)&h1p[lm * 34 + 4 * kc + kb];  // A-layout gather
            acc = wmma4(a2, B2[kc], acc);
        }
#pragma unroll
        for (int i = 0; i < 8; ++i) acc[i] = fast_tanh(acc[i]);

        // scatter h2 (stride 20)
#pragma unroll
        for (int v = 0; v < 8; ++v) h2p[(v + 8 * hi) * 20 + lm] = acc[v];

        // ---- layer 3 + maxwell + stress (lane m = lm, lanes 0-15) ----
        const int buf = t & 1;
        if (!hi) {
            const float4* row = (const float4*)&h2p[lm * 20];
            float corr = b3v;
#pragma unroll
            for (int j = 0; j < 4; ++j) {
                float4 r = row[j];
                corr = fmaf(r.x, w3v[4 * j + 0], corr);
                corr = fmaf(r.y, w3v[4 * j + 1], corr);
                corr = fmaf(r.z, w3v[4 * j + 2], corr);
                corr = fmaf(r.w, w3v[4 * j + 3], corr);
            }
            float mw = E * xc.x + eta * xc.y;
            sbuf[buf][n][lm] = (mw + corr) * iw;
        }
        __syncthreads();

        // next-step carry: lanes 16-31 need stress of batch row lm
        prev = sbuf[buf][n][lm];

        // ---- fusion head for batch (b0 + n); halves cover p = 2*hi + pp ----
#pragma unroll
        for (int pp = 0; pp < 2; ++pp) {
            int p = 2 * hi + pp;
            float v0 = sbuf[buf][0 * 4 + p][n] * cl[0];
            float v1 = sbuf[buf][1 * 4 + p][n] * cl[1];
            float v2 = sbuf[buf][2 * 4 + p][n] * cl[2];
            float v3 = sbuf[buf][3 * 4 + p][n] * cl[3];
            float hf = fmaf(v0, wf1v[0], bf1v);
            hf = fmaf(v1, wf1v[1], hf);
            hf = fmaf(v2, wf1v[2], hf);
            hf = fmaf(v3, wf1v[3], hf);
            hf = fmaxf(hf, 0.0f);
            float c = wf2v * hf;                 // lane lm holds feature f = lm
            c += __shfl_xor(c, 1);
            c += __shfl_xor(c, 2);
            c += __shfl_xor(c, 4);
            c += __shfl_xor(c, 8);
            if (lm == 0)
                out[(size_t)(p * BB + b0 + n) * TT + t] = c + bf2v;
        }
        xc = xn;
    }
}

extern "C" void kernel_launch(void* const* d_in, const int* in_sizes, int n_in,
                              void* d_out, int out_size, void* d_ws, size_t ws_size,
                              hipStream_t stream) {
    (void)in_sizes; (void)n_in; (void)d_ws; (void)ws_size; (void)out_size;
    pinn_kernel<<<dim3(16), dim3(512), 0, stream>>>(
        (const float*)d_in[0],  (const float*)d_in[1],  (const float*)d_in[2],
        (const float*)d_in[3],  (const float*)d_in[4],  (const float*)d_in[5],
        (const float*)d_in[6],  (const float*)d_in[7],  (const float*)d_in[8],
        (const float*)d_in[9],  (const float*)d_in[10], (const float*)d_in[11],
        (const float*)d_in[12], (const float*)d_in[13], (const float*)d_in[14],
        (float*)d_out);
}